// TNEP_14680198217892
// MI455X (gfx1250) — compile-verified
//
#include <hip/hip_runtime.h>
#include <hip/hip_bf16.h>

#define A_ATOMS 4096
#define MNEI 64
#define DQ 64
#define HID 128

typedef float v2f __attribute__((ext_vector_type(2)));
typedef float v8f __attribute__((ext_vector_type(8)));

// One block (128 threads = 4 wave32) per atom.
// Phase A: tiny type-routed MLP -> de_dq[64]; min-image rij^2*mask -> w[64].
// Phase B: dipole partial[c] = w^T * G_c * de_dq via V_WMMA_F32_16X16X4_F32,
//          streaming gradients exactly once (non-temporal).
__global__ __launch_bounds__(128) void tnep_atom_kernel(
    const float* __restrict__ desc,     // [A,64]
    const float* __restrict__ grads,    // [A,64,3,64]
    const int*   __restrict__ gidx,     // [A,64]
    const float* __restrict__ pos,      // [A,3]
    const int*   __restrict__ Zt,       // [A]
    const float* __restrict__ box,      // [3,3]
    const float* __restrict__ amask,    // [A]
    const float* __restrict__ nmask,    // [A,64]
    const float* __restrict__ W0,       // [8,64,128]
    const float* __restrict__ b0,       // [8,128]
    const float* __restrict__ W1,       // [8,128]
    float* __restrict__ part)           // [A,3] per-atom partial dipole (negated)
{
    __shared__ float s_desc[DQ];
    __shared__ float s_deda[HID];
    __shared__ float s_dedq[DQ];
    __shared__ float s_w[MNEI];
    __shared__ float s_box[9];
    __shared__ float s_ib[9];
    __shared__ float s_part[4][3];

    const int n = blockIdx.x;
    const int t = threadIdx.x;

    if (t < DQ) s_desc[t] = desc[n * DQ + t];
    if (t < 9)  s_box[t]  = box[t];
    __syncthreads();

    if (t == 0) {
        // 3x3 inverse (cofactors), row-major
        float b00 = s_box[0], b01 = s_box[1], b02 = s_box[2];
        float b10 = s_box[3], b11 = s_box[4], b12 = s_box[5];
        float b20 = s_box[6], b21 = s_box[7], b22 = s_box[8];
        float det = b00 * (b11 * b22 - b12 * b21)
                  - b01 * (b10 * b22 - b12 * b20)
                  + b02 * (b10 * b21 - b11 * b20);
        float id = 1.0f / det;
        s_ib[0] =  (b11 * b22 - b12 * b21) * id;
        s_ib[1] = -(b01 * b22 - b02 * b21) * id;
        s_ib[2] =  (b01 * b12 - b02 * b11) * id;
        s_ib[3] = -(b10 * b22 - b12 * b20) * id;
        s_ib[4] =  (b00 * b22 - b02 * b20) * id;
        s_ib[5] = -(b00 * b12 - b02 * b10) * id;
        s_ib[6] =  (b10 * b21 - b11 * b20) * id;
        s_ib[7] = -(b00 * b21 - b01 * b20) * id;
        s_ib[8] =  (b00 * b11 - b01 * b10) * id;
    }

    const int z = Zt[n];
    const float am = amask[n];
    const float* __restrict__ W0z = W0 + z * DQ * HID;

    // h[t] = tanh(desc . W0z[:,t] + b0) * amask ; de_da[t] = (1-h^2)*W1[t]
    {
        float acc = b0[z * HID + t];
        for (int q = 0; q < DQ; ++q)
            acc = fmaf(s_desc[q], W0z[q * HID + t], acc);
        float th = tanhf(acc) * am;
        float dt = 1.0f - th * th;
        s_deda[t] = dt * W1[z * HID + t];
    }
    __syncthreads();

    if (t < DQ) {
        // de_dq[q] = sum_h de_da[h] * W0z[q,h]
        const float* __restrict__ row = W0z + t * HID;
        float acc = 0.0f;
        for (int h = 0; h < HID; ++h)
            acc = fmaf(s_deda[h], row[h], acc);
        s_dedq[t] = acc;
    } else {
        // w[m] = |min_image(r_j - r_i)|^2 * neighbor_mask
        const int m = t - DQ;
        const int j = gidx[n * MNEI + m];
        float dx = pos[j * 3 + 0] - pos[n * 3 + 0];
        float dy = pos[j * 3 + 1] - pos[n * 3 + 1];
        float dz = pos[j * 3 + 2] - pos[n * 3 + 2];
        // frac = dr @ inv_box (row-vector x matrix)
        float fx = dx * s_ib[0] + dy * s_ib[3] + dz * s_ib[6];
        float fy = dx * s_ib[1] + dy * s_ib[4] + dz * s_ib[7];
        float fz = dx * s_ib[2] + dy * s_ib[5] + dz * s_ib[8];
        fx -= rintf(fx); fy -= rintf(fy); fz -= rintf(fz);  // RNE like jnp.round
        float gx = fx * s_box[0] + fy * s_box[3] + fz * s_box[6];
        float gy = fx * s_box[1] + fy * s_box[4] + fz * s_box[7];
        float gz = fx * s_box[2] + fy * s_box[5] + fz * s_box[8];
        float r2 = gx * gx + gy * gy + gz * gz;
        s_w[m] = r2 * nmask[n * MNEI + m];
    }
    __syncthreads();

    // Phase B: s[c] = sum_mt w_tile^T * (G_c tile . de_dq) via f32 WMMA.
    const int wv    = t >> 5;        // wave 0..3
    const int lane  = t & 31;
    const int lhalf = lane >> 4;     // 0: K=0,1 / rows 0..7 ; 1: K=2,3 / rows 8..15
    const int lmod  = lane & 15;
    const float* __restrict__ Gatom = grads + (size_t)n * (MNEI * 3 * DQ);

    for (int tid = wv; tid < 12; tid += 4) {   // uniform per wave -> EXEC all 1s
        const int c  = tid >> 2;               // 0..2
        const int mt = tid & 3;                // 0..3
        // A-fragment row pointer: lane holds M = 16*mt + lmod, K = 2*lhalf + {0,1}
        const float* __restrict__ rowp =
            Gatom + (16 * mt + lmod) * (3 * DQ) + c * DQ + 2 * lhalf;
        v8f dacc = {};
        #pragma unroll
        for (int kt = 0; kt < 16; ++kt) {
            v2f a = __builtin_nontemporal_load((const v2f*)(rowp + 4 * kt));
            v2f b;
            b.x = s_dedq[4 * kt + 2 * lhalf + 0];   // B replicated across 16 cols
            b.y = s_dedq[4 * kt + 2 * lhalf + 1];
            dacc = __builtin_amdgcn_wmma_f32_16x16x4_f32(
                false, a, false, b, (short)0, dacc, false, false);
        }
        // D vgpr r holds t[16*mt + r + 8*lhalf] (identical across 16 cols)
        float p = 0.0f;
        const int mbase = 16 * mt + 8 * lhalf;
        #pragma unroll
        for (int r = 0; r < 8; ++r)
            p = fmaf(s_w[mbase + r], dacc[r], p);
        float tv = __shfl(p, 0, 32) + __shfl(p, 16, 32); // rows 0..7 + rows 8..15
        if (lane == 0) s_part[wv][c] = tv;
    }
    __syncthreads();

    if (t < 3) {
        float s = s_part[0][t] + s_part[1][t] + s_part[2][t] + s_part[3][t];
        part[n * 3 + t] = -s;
    }
}

// Deterministic fixed-tree reduction of per-atom partials -> dipole[3].
__global__ __launch_bounds__(256) void tnep_reduce_kernel(
    const float* __restrict__ part, float* __restrict__ out)
{
    __shared__ float sm[256 * 3];
    const int t = threadIdx.x;
    float a0 = 0.0f, a1 = 0.0f, a2 = 0.0f;
    for (int n = t; n < A_ATOMS; n += 256) {
        a0 += part[n * 3 + 0];
        a1 += part[n * 3 + 1];
        a2 += part[n * 3 + 2];
    }
    sm[t * 3 + 0] = a0; sm[t * 3 + 1] = a1; sm[t * 3 + 2] = a2;
    __syncthreads();
    for (int s = 128; s > 0; s >>= 1) {
        if (t < s) {
            sm[t * 3 + 0] += sm[(t + s) * 3 + 0];
            sm[t * 3 + 1] += sm[(t + s) * 3 + 1];
            sm[t * 3 + 2] += sm[(t + s) * 3 + 2];
        }
        __syncthreads();
    }
    if (t < 3) out[t] = sm[t];
}

extern "C" void kernel_launch(void* const* d_in, const int* in_sizes, int n_in,
                              void* d_out, int out_size, void* d_ws, size_t ws_size,
                              hipStream_t stream) {
    (void)in_sizes; (void)n_in; (void)out_size; (void)ws_size;
    const float* desc  = (const float*)d_in[0];
    const float* grads = (const float*)d_in[1];
    const int*   gidx  = (const int*)  d_in[2];
    const float* pos   = (const float*)d_in[3];
    const int*   Zt    = (const int*)  d_in[4];
    const float* box   = (const float*)d_in[5];
    const float* amask = (const float*)d_in[6];
    const float* nmask = (const float*)d_in[7];
    const float* W0    = (const float*)d_in[8];
    const float* b0    = (const float*)d_in[9];
    const float* W1    = (const float*)d_in[10];
    // d_in[11] = b1 : drops out of force derivative, unused.

    float* part = (float*)d_ws;          // [A,3] fully overwritten each call
    float* out  = (float*)d_out;         // [3]

    tnep_atom_kernel<<<dim3(A_ATOMS), dim3(128), 0, stream>>>(
        desc, grads, gidx, pos, Zt, box, amask, nmask, W0, b0, W1, part);
    tnep_reduce_kernel<<<dim3(1), dim3(256), 0, stream>>>(part, out);
}